// LinearAttention_Cross_11269994184767
// MI455X (gfx1250) — compile-verified
//
#include <hip/hip_runtime.h>
#include <hip/hip_bf16.h>

// Problem constants (fixed by the reference setup).
#define NB 4
#define LQ 2048
#define SK 2048
#define NH 16
#define ED 64                 // E == D == 64
#define ROWSTRIDE (NH * ED)   // stride between consecutive s/l rows = 1024 floats
#define SPLIT 8               // S-split blocks per (n,h) in the reduction pass
#define EPS_F 1e-6f

typedef float v2f __attribute__((ext_vector_type(2)));
typedef float v8f __attribute__((ext_vector_type(8)));
typedef unsigned int u32x4 __attribute__((ext_vector_type(4)));
typedef int i32x4 __attribute__((ext_vector_type(4)));
typedef int i32x8 __attribute__((ext_vector_type(8)));

#if __has_builtin(__builtin_amdgcn_tensor_load_to_lds) && \
    __has_builtin(__builtin_amdgcn_s_wait_tensorcnt)
#define USE_TDM 1
#else
#define USE_TDM 0
#endif

__device__ __forceinline__ float phi(float x) {
    // elu(x) + 1  ==  x+1 (x>0)  |  exp(x) (x<=0)
    return x > 0.0f ? x + 1.0f : __expf(x);
}

__device__ __forceinline__ v8f wmma4(v2f a, v2f b, v8f c) {
    // D = A(16x4 f32) x B(4x16 f32) + C(16x16 f32), full fp32 precision.
    return __builtin_amdgcn_wmma_f32_16x16x4_f32(
        /*neg_a=*/false, a, /*neg_b=*/false, b,
        /*c_mod=*/(short)0, c, /*reuse_a=*/false, /*reuse_b=*/false);
}

__global__ void zero_ws_kernel(float* __restrict__ p, int n) {
    int i = blockIdx.x * blockDim.x + threadIdx.x;
    if (i < n) p[i] = 0.0f;
}

// Pass 1: per (n,h): M[e,m] = sum_s phi(K[s,e])*kl[s]*V[s,m];  Ksum[e] = sum_s phi(K[s,e])*kl[s]
__global__ __launch_bounds__(256, 1)
void kv_reduce_kernel(const float* __restrict__ K, const float* __restrict__ V,
                      const float* __restrict__ kl,
                      float* __restrict__ kvOut, float* __restrict__ ksOut) {
    __shared__ float ldsKV[ED * ED];
    __shared__ float ldsKs[ED];

    const int bi    = blockIdx.x;
    const int split = bi % SPLIT;
    const int nh    = bi / SPLIT;
    const int n     = nh / NH;
    const int h     = nh % NH;
    const int tid   = threadIdx.x;
    const int wave  = tid >> 5;
    const int lane  = tid & 31;
    const int half  = lane & 15;
    const bool hi   = (lane >= 16);

    v8f acc[16];
    #pragma unroll
    for (int i = 0; i < 16; ++i) acc[i] = (v8f)0.0f;
    float kpart[4] = {0.f, 0.f, 0.f, 0.f};

    const int sBlock   = split * (SK / SPLIT);   // 256 s-rows per block
    const float* kl_n  = kl + n * SK;

    for (int it = 0; it < (SK / SPLIT) / 32; ++it) {     // 8 iterations
        const int s0   = sBlock + it * 32 + wave * 4;    // this wave's 4-row chunk
        const int srow = s0 + (hi ? 2 : 0);              // A-layout: lanes 16-31 hold K=2,3
        const float klv0 = kl_n[srow];
        const float klv1 = kl_n[srow + 1];
        const float* kr0 = K + (size_t)(n * SK + srow) * ROWSTRIDE + h * ED;
        const float* kr1 = kr0 + ROWSTRIDE;
        const float* vr0 = V + (size_t)(n * SK + srow) * ROWSTRIDE + h * ED;
        const float* vr1 = vr0 + ROWSTRIDE;

        // Prefetch the next chunk's rows (32 s ahead) -> global_prefetch_b8.
        __builtin_prefetch(kr0 + 32 * ROWSTRIDE, 0, 0);
        __builtin_prefetch(vr0 + 32 * ROWSTRIDE, 0, 0);

        v2f afrag[4], bfrag[4];
        #pragma unroll
        for (int t = 0; t < 4; ++t) {
            const int e = t * 16 + half;
            const float ax = phi(kr0[e]) * klv0;
            const float ay = phi(kr1[e]) * klv1;
            v2f a; a.x = ax; a.y = ay;
            afrag[t] = a;
            kpart[t] += ax + ay;             // each K element loaded exactly once
            v2f b; b.x = vr0[e]; b.y = vr1[e];
            bfrag[t] = b;
        }
        #pragma unroll
        for (int et = 0; et < 4; ++et)
            #pragma unroll
            for (int mt = 0; mt < 4; ++mt)
                acc[et * 4 + mt] = wmma4(afrag[et], bfrag[mt], acc[et * 4 + mt]);
    }

    // Cross-wave reduce via LDS atomics (ds_add_f32).
    for (int i = tid; i < ED * ED; i += 256) ldsKV[i] = 0.0f;
    if (tid < ED) ldsKs[tid] = 0.0f;
    __syncthreads();
    #pragma unroll
    for (int et = 0; et < 4; ++et) {
        #pragma unroll
        for (int mt = 0; mt < 4; ++mt) {
            #pragma unroll
            for (int r = 0; r < 8; ++r) {
                const int e = et * 16 + r + (hi ? 8 : 0);  // C/D layout: M=r / r+8
                const int m = mt * 16 + half;
                atomicAdd(&ldsKV[e * ED + m], acc[et * 4 + mt][r]);
            }
        }
    }
    #pragma unroll
    for (int t = 0; t < 4; ++t) atomicAdd(&ldsKs[t * 16 + half], kpart[t]);
    __syncthreads();

    // Cross-block reduce via global fp32 atomics.
    float* kvDst = kvOut + (size_t)nh * ED * ED;
    for (int i = tid; i < ED * ED; i += 256) atomicAdd(&kvDst[i], ldsKV[i]);
    if (tid < ED) atomicAdd(&ksOut[nh * ED + tid], ldsKs[tid]);
}

// Pass 2: out[l,m] = z(l) * sum_e phi(Q[l,e]) * M[e,m],  z = 1/(phi(Q[l,:])·Ksum + eps)
__global__ __launch_bounds__(256, 1)
void apply_kernel(const float* __restrict__ Q, const float* __restrict__ KV,
                  const float* __restrict__ Ks, float* __restrict__ Out) {
    __shared__ float Mlds[ED * ED];
    __shared__ float kslds[ED];

    const int LB   = LQ / 128;          // 16 l-blocks of 128 rows
    const int bi   = blockIdx.x;
    const int lb   = bi % LB;
    const int nh   = bi / LB;
    const int n    = nh / NH;
    const int h    = nh % NH;
    const int tid  = threadIdx.x;
    const int wave = tid >> 5;
    const int lane = tid & 31;
    const int half = lane & 15;
    const bool hi  = (lane >= 16);

#if USE_TDM
    // Stage M (64x64 f32, 16KB contiguous 2D tile) into LDS with one Tensor Data
    // Mover DMA. TDM ignores EXEC, so exactly one wave issues the descriptor;
    // it waits on TENSORcnt, then the workgroup barrier publishes the LDS data.
    if (wave == 0) {
        const unsigned ldsAddr = (unsigned)(uintptr_t)(&Mlds[0]);   // low 32b = LDS offset
        const unsigned long long ga =
            (unsigned long long)(uintptr_t)(KV + (size_t)nh * ED * ED);
        u32x4 g0;
        g0[0] = 1u;                                   // count=1, user descriptor
        g0[1] = ldsAddr;                              // lds_addr (bytes)
        g0[2] = (unsigned)(ga & 0xFFFFFFFFu);         // global_addr[31:0] (bits 95:64)
        g0[3] = (unsigned)((ga >> 32) & 0x01FFFFFFu)  // global_addr[56:32]
              | (2u << 30);                           // type = 2 ("image")
        i32x8 g1;
        g1[0] = (2 << 16);          // workgroup_mask=0, data_size=2 (4 bytes)
        g1[1] = (ED << 16);         // tensor_dim0[15:0] at bits 63:48
        g1[2] = (ED << 16);         // tensor_dim0 hi=0 | tensor_dim1[15:0] at 95:80
        g1[3] = (ED << 16);         // tensor_dim1 hi=0 | tile_dim0=64 at 127:112
        g1[4] = ED;                 // tile_dim1=64, tile_dim2=0
        g1[5] = ED;                 // tensor_dim0_stride low32 = 64
        g1[6] = 0;                  // stride0 hi16 = 0 | tensor_dim1_stride lo = 0 (2D)
        g1[7] = 0;
        i32x4 gz; gz[0] = 0; gz[1] = 0; gz[2] = 0; gz[3] = 0;   // groups 2/3 unused (2D)
#if __clang_major__ >= 23
        __builtin_amdgcn_tensor_load_to_lds(g0, g1, gz, gz, (i32x8)0, 0);
#else
        __builtin_amdgcn_tensor_load_to_lds(g0, g1, gz, gz, 0);
#endif
        __builtin_amdgcn_s_wait_tensorcnt(0);
    }
#else
    for (int i = tid; i < ED * ED; i += 256) Mlds[i] = KV[(size_t)nh * ED * ED + i];
#endif
    if (tid < ED) kslds[tid] = Ks[nh * ED + tid];
    __syncthreads();

    const int ltile = lb * 128 + wave * 16;
    const int row   = ltile + half;                     // A-layout: M = lane&15
    const float* qrow = Q + (size_t)(n * LQ + row) * ROWSTRIDE + h * ED;

    v8f acc[4];
    #pragma unroll
    for (int mt = 0; mt < 4; ++mt) acc[mt] = (v8f)0.0f;
    float zpart = 0.0f;

    #pragma unroll
    for (int k = 0; k < 16; ++k) {
        const int e = 4 * k + (hi ? 2 : 0);             // K=0,1 (lo half) / K=2,3 (hi half)
        const float2 q2 = *reinterpret_cast<const float2*>(qrow + e);
        const float qx = phi(q2.x);
        const float qy = phi(q2.y);
        zpart += qx * kslds[e] + qy * kslds[e + 1];
        v2f a; a.x = qx; a.y = qy;
        #pragma unroll
        for (int mt = 0; mt < 4; ++mt) {
            v2f b;
            b.x = Mlds[e * ED + mt * 16 + half];
            b.y = Mlds[(e + 1) * ED + mt * 16 + half];
            acc[mt] = wmma4(a, b, acc[mt]);
        }
    }

    // Combine the two lane-halves of the normalizer, then broadcast per output row.
    const float zsum = zpart + __shfl_xor(zpart, 16, 32);
    const float z = 1.0f / (zsum + EPS_F);
    float zr[8];
    #pragma unroll
    for (int r = 0; r < 8; ++r)
        zr[r] = __shfl(z, r + ((lane & 16) >> 1), 32);  // row = ltile + r (+8 for hi half)

    #pragma unroll
    for (int mt = 0; mt < 4; ++mt) {
        #pragma unroll
        for (int r = 0; r < 8; ++r) {
            const int l = ltile + r + (hi ? 8 : 0);
            Out[(size_t)(n * LQ + l) * ROWSTRIDE + h * ED + mt * 16 + half] =
                acc[mt][r] * zr[r];
        }
    }
}

extern "C" void kernel_launch(void* const* d_in, const int* in_sizes, int n_in,
                              void* d_out, int out_size, void* d_ws, size_t ws_size,
                              hipStream_t stream) {
    (void)in_sizes; (void)n_in; (void)out_size; (void)ws_size;
    const float* q1  = (const float*)d_in[0];
    const float* k1  = (const float*)d_in[1];
    const float* v1  = (const float*)d_in[2];
    const float* q2  = (const float*)d_in[3];
    const float* k2  = (const float*)d_in[4];
    const float* v2  = (const float*)d_in[5];
    const float* kl1 = (const float*)d_in[6];
    const float* kl2 = (const float*)d_in[7];
    float* out = (float*)d_out;

    float* ws  = (float*)d_ws;
    const int kvElems = NB * NH * ED * ED;   // 262144 per stream
    const int ksElems = NB * NH * ED;        // 4096 per stream
    float* kvA = ws;                         // KV from stream 1 (keys1/values1)
    float* kvB = ws + kvElems;               // KV from stream 2
    float* ksA = kvB + kvElems;
    float* ksB = ksA + ksElems;
    const int totalWs = 2 * kvElems + 2 * ksElems;

    zero_ws_kernel<<<(totalWs + 255) / 256, 256, 0, stream>>>(ws, totalWs);

    kv_reduce_kernel<<<NB * NH * SPLIT, 256, 0, stream>>>(k1, v1, kl1, kvA, ksA);
    kv_reduce_kernel<<<NB * NH * SPLIT, 256, 0, stream>>>(k2, v2, kl2, kvB, ksB);

    const int LB = LQ / 128;
    // V1 = Q1 attending to stream-2 KV; V2 = Q2 attending to stream-1 KV.
    apply_kernel<<<NB * NH * LB, 256, 0, stream>>>(q1, kvB, ksB, out);
    apply_kernel<<<NB * NH * LB, 256, 0, stream>>>(q2, kvA, ksA,
                                                   out + (size_t)NB * LQ * NH * ED);
}